// SteinModel_82420422410495
// MI455X (gfx1250) — compile-verified
//
#include <hip/hip_runtime.h>
#include <math.h>

// SVGD gradient correction, algebraically reduced:
//   G   = var @ grad^T                      [128,128]
//   S   = G @ G^T / n                       [128,128]   (since M = grad^T grad / n is symmetric)
//   dMsd[i,j] = S[i,i] + S[j,j] - 2 S[i,j]
//   h   = median(dMsd)^2 / log(n)   (percentile 50, 'nearest' -> index 8192 of 16384)
//   kern = exp(-dMsd / 2h)
//   vM  = G @ grad / n                      [128,2048]  (= var @ M, M never materialized)
//   out = kern @ (grad + vM/h) - (rowsum(kern)/h) * vM
// ~270 MFLOP total, ~3 MB traffic -> latency bound; use f32 WMMA (16x16x4) for full
// reference precision while driving the matrix pipes.

#define N_P 128
#define DIM 2048

typedef float v2f __attribute__((ext_vector_type(2)));
typedef float v8f __attribute__((ext_vector_type(8)));

static __device__ __forceinline__ v8f wmma_f32(v2f a, v2f b, v8f c) {
  // D = A(16x4) * B(4x16) + C(16x16), all f32.
  return __builtin_amdgcn_wmma_f32_16x16x4_f32(false, a, false, b, (short)0, c,
                                               false, false);
}

// ---------------- Kernel 1: G = var @ grad^T  (64 tiles, 1 wave/tile, K=2048)
__global__ void __launch_bounds__(256) k_gram(const float* __restrict__ var,
                                              const float* __restrict__ grad,
                                              float* __restrict__ G) {
  const int wave = (blockIdx.x << 3) | (threadIdx.x >> 5);  // 0..63
  const int lane = threadIdx.x & 31;
  const int tm = (wave >> 3) << 4;
  const int tn = (wave & 7) << 4;
  const int r = lane & 15;
  const int koff = (lane >> 4) << 1;  // 0 or 2
  const float* arow = var + (tm + r) * DIM + koff;
  const float* brow = grad + (tn + r) * DIM + koff;  // B col j = grad row j
  v8f c = {};
#pragma unroll 4
  for (int k = 0; k < DIM; k += 4) {
    v2f a = *(const v2f*)(arow + k);
    v2f b = *(const v2f*)(brow + k);
    c = wmma_f32(a, b, c);
  }
  const int rbase = tm + ((lane >> 4) << 3);
#pragma unroll
  for (int v = 0; v < 8; ++v)
    G[(rbase + v) * N_P + (tn + r)] = c[v];
}

// ---------------- Kernel 2: S, dMsd, median (bit bisection), h, kern, rowsum.
// Single 1024-thread workgroup; dMsd kept entirely in LDS (64 KB << 320 KB).
__global__ void __launch_bounds__(1024) k_kern(const float* __restrict__ G,
                                               float* __restrict__ kern_g,
                                               float* __restrict__ rvec,
                                               float* __restrict__ invh_g) {
  __shared__ float Ssh[N_P * N_P];
  __shared__ float qsh[N_P];
  __shared__ unsigned cnt;
  const int tid = threadIdx.x;
  const int wv = tid >> 5;  // 0..31, each wave does tiles wv and wv+32
  const int lane = tid & 31;
  const int r = lane & 15;
  const int koff = (lane >> 4) << 1;
#pragma unroll
  for (int tt = 0; tt < 2; ++tt) {
    const int t = wv + (tt << 5);
    const int tm = (t >> 3) << 4;
    const int tn = (t & 7) << 4;
    const float* arow = G + (tm + r) * N_P + koff;
    const float* brow = G + (tn + r) * N_P + koff;  // S = G G^T: B col j = G row j
    v8f c = {};
#pragma unroll 4
    for (int k = 0; k < N_P; k += 4) {
      v2f a = *(const v2f*)(arow + k);
      v2f b = *(const v2f*)(brow + k);
      c = wmma_f32(a, b, c);
    }
    const int rbase = tm + ((lane >> 4) << 3);
#pragma unroll
    for (int v = 0; v < 8; ++v)
      Ssh[(rbase + v) * N_P + (tn + r)] = c[v] * (1.0f / N_P);
  }
  __syncthreads();
  if (tid < N_P) qsh[tid] = Ssh[tid * N_P + tid];
  __syncthreads();
  // dMsd in place; clamp rounding residue so uint-bisection stays monotone
#pragma unroll
  for (int t = 0; t < 16; ++t) {
    const int idx = (tid << 4) + t;
    const int i = idx >> 7, j = idx & (N_P - 1);
    Ssh[idx] = fmaxf(qsh[i] + qsh[j] - 2.0f * Ssh[idx], 0.0f);
  }
  __syncthreads();
  // k-th order statistic (k=8192 zero-based; threshold 8193) via 32-step
  // binary search on non-negative float bit patterns.
  unsigned lo = 0u, hi = 0xFFFFFFFFu;
  for (int it = 0; it < 32; ++it) {
    const unsigned mid = lo + ((hi - lo) >> 1);
    if (tid == 0) cnt = 0u;
    __syncthreads();
    unsigned c = 0;
#pragma unroll
    for (int t = 0; t < 16; ++t)
      if (__float_as_uint(Ssh[(tid << 4) + t]) <= mid) ++c;
    atomicAdd(&cnt, c);
    __syncthreads();
    const unsigned total = cnt;
    __syncthreads();
    if (total >= 8193u) hi = mid; else lo = mid + 1u;
  }
  const float med = __uint_as_float(hi);
  const float h = med * med / logf((float)N_P);
  const float invh = 1.0f / h;
  const float inv2h = 0.5f * invh;
#pragma unroll
  for (int t = 0; t < 16; ++t) {
    const int idx = (tid << 4) + t;
    const float kv = expf(-Ssh[idx] * inv2h);
    Ssh[idx] = kv;
    kern_g[idx] = kv;
  }
  __syncthreads();
  if (tid < N_P) {
    float s = 0.0f;
    for (int j = 0; j < N_P; ++j) s += Ssh[tid * N_P + j];
    rvec[tid] = s * invh;
  }
  if (tid == 0) invh_g[0] = invh;
}

// ---------------- Kernel 3: vM = G @ grad / n   (8 x 128 tiles, 1 wave/tile)
__global__ void __launch_bounds__(256) k_vm(const float* __restrict__ G,
                                            const float* __restrict__ grad,
                                            float* __restrict__ vM) {
  const int wave = (blockIdx.x << 3) | (threadIdx.x >> 5);  // 0..1023
  const int lane = threadIdx.x & 31;
  const int ti = wave >> 7;
  const int td = wave & 127;
  const int r = lane & 15;
  const int koff = (lane >> 4) << 1;
  const float* arow = G + (ti * 16 + r) * N_P + koff;
  const int col = td * 16 + r;
  v8f c = {};
#pragma unroll 4
  for (int k = 0; k < N_P; k += 4) {
    v2f a = *(const v2f*)(arow + k);
    const int k0 = k + koff;
    v2f b;
    b.x = grad[k0 * DIM + col];
    b.y = grad[(k0 + 1) * DIM + col];
    c = wmma_f32(a, b, c);
  }
  const int rbase = ti * 16 + ((lane >> 4) << 3);
#pragma unroll
  for (int v = 0; v < 8; ++v)
    vM[(rbase + v) * DIM + col] = c[v] * (1.0f / N_P);
}

// ---------------- Kernel 4: out = kern @ (grad + vM/h) - (rowsum/h)*vM
__global__ void __launch_bounds__(256) k_out(const float* __restrict__ kern,
                                             const float* __restrict__ grad,
                                             const float* __restrict__ vM,
                                             const float* __restrict__ rvec,
                                             const float* __restrict__ invh_g,
                                             float* __restrict__ out) {
  const int wave = (blockIdx.x << 3) | (threadIdx.x >> 5);
  const int lane = threadIdx.x & 31;
  const int ti = wave >> 7;
  const int td = wave & 127;
  const int r = lane & 15;
  const int koff = (lane >> 4) << 1;
  const float invh = invh_g[0];
  const float* arow = kern + (ti * 16 + r) * N_P + koff;
  const int col = td * 16 + r;
  v8f c = {};
#pragma unroll 4
  for (int k = 0; k < N_P; k += 4) {
    v2f a = *(const v2f*)(arow + k);
    const int k0 = k + koff;
    v2f b;
    b.x = fmaf(vM[k0 * DIM + col], invh, grad[k0 * DIM + col]);
    b.y = fmaf(vM[(k0 + 1) * DIM + col], invh, grad[(k0 + 1) * DIM + col]);
    c = wmma_f32(a, b, c);
  }
  const int rbase = ti * 16 + ((lane >> 4) << 3);
#pragma unroll
  for (int v = 0; v < 8; ++v) {
    const int row = rbase + v;
    out[row * DIM + col] = c[v] - rvec[row] * vM[row * DIM + col];
  }
}

extern "C" void kernel_launch(void* const* d_in, const int* in_sizes, int n_in,
                              void* d_out, int out_size, void* d_ws,
                              size_t ws_size, hipStream_t stream) {
  (void)in_sizes; (void)n_in; (void)out_size; (void)ws_size;
  const float* var = (const float*)d_in[0];
  const float* grad = (const float*)d_in[1];
  float* out = (float*)d_out;
  float* ws = (float*)d_ws;
  // workspace layout (floats): G[16384] | kern[16384] | rvec[128] | invh[1] | pad | vM[262144]
  float* G = ws;
  float* kern = ws + 16384;
  float* rvec = ws + 32768;
  float* invh = ws + 32896;
  float* vM = ws + 32904;  // byte offset 131616, 8B aligned

  k_gram<<<8, 256, 0, stream>>>(var, grad, G);
  k_kern<<<1, 1024, 0, stream>>>(G, kern, rvec, invh);
  k_vm<<<128, 256, 0, stream>>>(G, grad, vM);
  k_out<<<128, 256, 0, stream>>>(kern, grad, vM, rvec, invh, out);
}